// GConv_51307679318309
// MI455X (gfx1250) — compile-verified
//
#include <hip/hip_runtime.h>

#define NN 50000
#define NE 800000
#define DD 256
#define LDSP 260           // padded LDS row pitch (floats): 260 % 64 = 4 -> conflict-free A reads
#define EPB 8              // edges per block in scatter

typedef __attribute__((ext_vector_type(2))) float v2f;
typedef __attribute__((ext_vector_type(8))) float v8f;

// ---------------------------------------------------------------------------
// deg[i] = 1.0 (self-loop weight)
__global__ void __launch_bounds__(256) k_deg_init(float* __restrict__ deg) {
    int i = blockIdx.x * 256 + threadIdx.x;
    if (i < NN) deg[i] = 1.0f;
}

// deg[col[e]] += ew[e]
__global__ void __launch_bounds__(256) k_deg_accum(const int* __restrict__ col,
                                                   const float* __restrict__ ew,
                                                   float* __restrict__ deg) {
    int e = blockIdx.x * 256 + threadIdx.x;
    if (e < NE) atomicAdd(deg + col[e], ew[e]);
}

// dinv[i] = rsqrt(deg[i])   (deg >= 1 always)
__global__ void __launch_bounds__(256) k_dinv(const float* __restrict__ deg,
                                              float* __restrict__ dinv) {
    int i = blockIdx.x * 256 + threadIdx.x;
    if (i < NN) dinv[i] = rsqrtf(deg[i]);
}

// zero a float buffer (float4 stores)
__global__ void __launch_bounds__(256) k_zero(float4* __restrict__ p, int n4) {
    int i = blockIdx.x * 256 + threadIdx.x;
    if (i < n4) p[i] = make_float4(0.f, 0.f, 0.f, 0.f);
}

// ---------------------------------------------------------------------------
// H[M=16 tile, all 256 cols] = X_tile @ W using V_WMMA_F32_16X16X4_F32.
// Block = 128 threads = 4 waves. blockIdx.x = M tile (3125), blockIdx.y = 64-col group (4).
// Wave w owns columns [blockIdx.y*64 + w*16, +16).
__global__ void __launch_bounds__(128)
k_gemm_wmma(const float* __restrict__ X, const float* __restrict__ W,
            float* __restrict__ H) {
    __shared__ float lds[16 * LDSP];
    const int row0 = blockIdx.x * 16;
    const int tid  = threadIdx.x;

    // cooperatively stage the 16x256 A tile into LDS (pitch-padded)
    for (int i = tid; i < 16 * (DD / 4); i += 128) {
        const int r  = i >> 6;
        const int c4 = (i & 63) << 2;
        const float4 v = *(const float4*)(X + (size_t)(row0 + r) * DD + c4);
        float* dst = &lds[r * LDSP + c4];
        dst[0] = v.x; dst[1] = v.y; dst[2] = v.z; dst[3] = v.w;
    }
    __syncthreads();

    const int lane = tid & 31;
    const int wave = tid >> 5;
    const int half = lane >> 4;   // 0: lanes 0-15, 1: lanes 16-31
    const int l16  = lane & 15;
    const int col0 = blockIdx.y * 64 + wave * 16;

    v8f c = {};
#if defined(__gfx1250__) && __has_builtin(__builtin_amdgcn_wmma_f32_16x16x4_f32)
    #pragma unroll 4
    for (int k = 0; k < DD; k += 4) {
        const int kk = k + 2 * half;
        // A 16x4 fp32 fragment (ISA 7.12.2): lanes<16 -> K={k,k+1}, lanes>=16 -> K={k+2,k+3}
        v2f a;
        a.x = lds[l16 * LDSP + kk];
        a.y = lds[l16 * LDSP + kk + 1];
        // B 4x16 fp32 fragment, row K striped across lanes (mirror of A)
        v2f b;
        b.x = W[(size_t)(kk)     * DD + col0 + l16];
        b.y = W[(size_t)(kk + 1) * DD + col0 + l16];
        c = __builtin_amdgcn_wmma_f32_16x16x4_f32(false, a, false, b,
                                                  (short)0, c, false, false);
    }
#else
    // scalar fallback producing the identical C-fragment layout
    for (int k = 0; k < DD; ++k) {
        const float bw = W[(size_t)k * DD + col0 + l16];
        #pragma unroll
        for (int j = 0; j < 8; ++j)
            c[j] += lds[(j + 8 * half) * LDSP + k] * bw;
    }
#endif
    // C/D layout: VGPR j -> row (j + 8*half), col = l16
    #pragma unroll
    for (int j = 0; j < 8; ++j) {
        const int r  = row0 + j + 8 * half;
        H[(size_t)r * DD + col0 + l16] = c[j];
    }
}

// ---------------------------------------------------------------------------
// agg[col[e], :] += (dinv[row]*ew*dinv[col]) * H[row[e], :]
// 256 threads: thread d handles feature d. L2-resident atomics (h/agg fit in 192MB L2).
__global__ void __launch_bounds__(256)
k_scatter(const float* __restrict__ H, const int* __restrict__ row,
          const int* __restrict__ col, const float* __restrict__ ew,
          const float* __restrict__ dinv, float* __restrict__ agg) {
    const int d    = threadIdx.x;
    const int base = blockIdx.x * EPB;
    #pragma unroll
    for (int i = 0; i < EPB; ++i) {
        const int e = base + i;
        if (e >= NE) return;
        const int r = row[e];
        const int t = col[e];
        const float norm = dinv[r] * ew[e] * dinv[t];
        atomicAdd(agg + (size_t)t * DD + d, norm * H[(size_t)r * DD + d]);
    }
}

// out[i,d] = relu(agg[i,d] + dinv[i]^2 * H[i,d] + b[d])
__global__ void __launch_bounds__(256)
k_combine_relu(const float* __restrict__ agg, const float* __restrict__ H,
               const float* __restrict__ dinv, const float* __restrict__ bias,
               float* __restrict__ out) {
    const int i = blockIdx.x;
    const int d = threadIdx.x;
    const float s = dinv[i] * dinv[i];
    const float v = agg[(size_t)i * DD + d] + s * H[(size_t)i * DD + d] + bias[d];
    out[(size_t)i * DD + d] = v > 0.f ? v : 0.f;
}

// ---------------------------------------------------------------------------
extern "C" void kernel_launch(void* const* d_in, const int* in_sizes, int n_in,
                              void* d_out, int out_size, void* d_ws, size_t ws_size,
                              hipStream_t stream) {
    const float* x  = (const float*)d_in[0];
    const int*   ei = (const int*)  d_in[1];
    const float* ew = (const float*)d_in[2];
    const float* W1 = (const float*)d_in[3];
    const float* b1 = (const float*)d_in[4];
    const float* W2 = (const float*)d_in[5];
    const float* b2 = (const float*)d_in[6];
    const int* row = ei;        // edge_index[0]
    const int* col = ei + NE;   // edge_index[1]
    float* out = (float*)d_out;

    // workspace layout
    float* bufH = (float*)d_ws;                         // N*D : h = in @ W
    float* bufA = bufH + (size_t)NN * DD;               // N*D : agg / z
    float* deg  = bufA + (size_t)NN * DD;               // N
    float* dinv = deg + NN;                             // N

    const int gN  = (NN + 255) / 256;
    const int gE  = (NE + 255) / 256;
    const int gZ  = ((NN * DD / 4) + 255) / 256;
    const int gSc = (NE + EPB - 1) / EPB;
    const dim3 gG(NN / 16, DD / 64);                    // (3125, 4)

    // shared normalization (graph is identical across layers)
    k_deg_init <<<gN, 256, 0, stream>>>(deg);
    k_deg_accum<<<gE, 256, 0, stream>>>(col, ew, deg);
    k_dinv     <<<gN, 256, 0, stream>>>(deg, dinv);

    // ---- layer 1 ----
    k_gemm_wmma<<<gG, 128, 0, stream>>>(x, W1, bufH);
    k_zero     <<<gZ, 256, 0, stream>>>((float4*)bufA, NN * DD / 4);
    k_scatter  <<<gSc, 256, 0, stream>>>(bufH, row, col, ew, dinv, bufA);
    k_combine_relu<<<NN, 256, 0, stream>>>(bufA, bufH, dinv, b1, bufA); // z1 -> bufA

    // ---- layer 2 ----
    k_gemm_wmma<<<gG, 128, 0, stream>>>(bufA, W2, bufH);
    k_zero     <<<gZ, 256, 0, stream>>>((float4*)bufA, NN * DD / 4);
    k_scatter  <<<gSc, 256, 0, stream>>>(bufH, row, col, ew, dinv, bufA);
    k_combine_relu<<<NN, 256, 0, stream>>>(bufA, bufH, dinv, b2, out);
}